// DualBlock_54640573939784
// MI455X (gfx1250) — compile-verified
//
#include <hip/hip_runtime.h>
#include <hip/hip_bf16.h>

typedef __attribute__((ext_vector_type(2))) float v2f;
typedef __attribute__((ext_vector_type(8))) float v8f;

#define NV_     12000
#define NF_     24000
#define C_IN_   64
#define F4_     128
#define G2_     64
#define C_OUT_  128
#define XSTRIDE 320
#define MAX_NN_ 16
#define K_PT_   10
#define K_MESH_ 9
#define MAX_NF_ 8
#define EPS_    1e-5f
#define R2_     0.0625f

// ---------------------------------------------------------------------------
// WMMA core: accumulate a 16-row x 16-col tile over a 64-deep K chunk.
// A tile lives in LDS (16x64 f32); W is (K,O) row-major, pointer pre-advanced
// to row k0. Uses V_WMMA_F32_16X16X4_F32 (ISA 7.12): A 16x4 = 2 VGPR/lane
// (lanes 0-15: K=0,1; lanes 16-31: K=2,3), D 16x16 = 8 VGPR.
// ---------------------------------------------------------------------------
__device__ __forceinline__ void wmma_chunk64(const float* __restrict__ A,
                                             const float* __restrict__ Wk,
                                             int O, int ncol0, v8f& acc)
{
  const int lane = threadIdx.x & 31;
  const int m    = lane & 15;
  const int koff = (lane >> 4) << 1;       // 0 for lanes 0-15, 2 for 16-31
  const int n    = ncol0 + (lane & 15);
#pragma unroll
  for (int kk = 0; kk < 64; kk += 4) {
    v2f a, b;
    a.x = A[m * 64 + kk + koff];
    a.y = A[m * 64 + kk + koff + 1];
    b.x = Wk[(size_t)(kk + koff) * O + n];
    b.y = Wk[(size_t)(kk + koff + 1) * O + n];
    acc = __builtin_amdgcn_wmma_f32_16x16x4_f32(false, a, false, b, (short)0,
                                                acc, false, false);
  }
}

// ---------------------------------------------------------------------------
// Scatter inputs (NV x 64) into the 320-wide concatenated feature buffer.
// ---------------------------------------------------------------------------
__global__ void __launch_bounds__(256) copy_in_kernel(
    const float* __restrict__ in, float* __restrict__ xbuf)
{
  int idx = blockIdx.x * blockDim.x + threadIdx.x;   // exact: NV*64 threads
  int r = idx >> 6, c = idx & 63;
  xbuf[(size_t)r * XSTRIDE + c] = in[idx];
}

// ---------------------------------------------------------------------------
// Point graph: per-point top-16 nearest (insertion sort, stable => earliest
// index wins ties like jax top_k), ball mask, SH-l2 coefficients.
// ---------------------------------------------------------------------------
__global__ void __launch_bounds__(128) point_graph_kernel(
    const float* __restrict__ vert, int* __restrict__ nnidx,
    int* __restrict__ nncnt, float* __restrict__ coeff)
{
  int i = blockIdx.x * blockDim.x + threadIdx.x;
  if (i >= NV_) return;
  float xi = vert[i * 3 + 0], yi = vert[i * 3 + 1], zi = vert[i * 3 + 2];
  float sqi = xi * xi + yi * yi + zi * zi;
  float dbest[MAX_NN_]; int ibest[MAX_NN_];
#pragma unroll
  for (int s = 0; s < MAX_NN_; ++s) { dbest[s] = 3.4e38f; ibest[s] = 0; }
  for (int j = 0; j < NV_; ++j) {
    float xj = vert[j * 3 + 0], yj = vert[j * 3 + 1], zj = vert[j * 3 + 2];
    float sqj = xj * xj + yj * yj + zj * zj;
    float d2 = sqi + sqj - 2.f * (xi * xj + yi * yj + zi * zj);
    if (d2 < dbest[MAX_NN_ - 1]) {
      int p = MAX_NN_ - 1;
      while (p > 0 && dbest[p - 1] > d2) {
        dbest[p] = dbest[p - 1]; ibest[p] = ibest[p - 1]; --p;
      }
      dbest[p] = d2; ibest[p] = j;
    }
  }
  int cnt = 0;
#pragma unroll
  for (int s = 0; s < MAX_NN_; ++s) {
    int j = ibest[s];
    float d2 = dbest[s] > 0.f ? dbest[s] : 0.f;
    bool inb = d2 < R2_;
    cnt += inb ? 1 : 0;
    nnidx[(size_t)i * MAX_NN_ + s] = j;
    float rx = vert[j * 3 + 0] - xi, ry = vert[j * 3 + 1] - yi,
          rz = vert[j * 3 + 2] - zi;
    float inv = 1.f / (sqrtf(d2) + 1e-12f);
    float ux = rx * inv, uy = ry * inv, uz = rz * inv;
    float ib = inb ? 1.f : 0.f;
    float ns = (j == i) ? 0.f : 1.f;
    float* cp = coeff + ((size_t)i * MAX_NN_ + s) * K_PT_;
    cp[0] = ((j == i) ? 1.f : 0.f) * ib;
    cp[1] = 0.28209479f * ns * ib;
    cp[2] = 0.48860251f * uy * ns * ib;
    cp[3] = 0.48860251f * uz * ns * ib;
    cp[4] = 0.48860251f * ux * ns * ib;
    cp[5] = 1.09254843f * ux * uy * ns * ib;
    cp[6] = 1.09254843f * uy * uz * ns * ib;
    cp[7] = 0.31539157f * (3.f * uz * uz - 1.f) * ns * ib;
    cp[8] = 1.09254843f * ux * uz * ns * ib;
    cp[9] = 0.54627421f * (ux * ux - uy * uy) * ns * ib;
  }
  nncnt[i] = cnt;
}

// ---------------------------------------------------------------------------
// v2f: y[f,o] = (1/3) * sum_{corner,c} X[face[f,corner],c] * W[corner*cin+c,o] + b
// O = 128 always. Gathers fused into LDS A-tile build.
// ---------------------------------------------------------------------------
__global__ void __launch_bounds__(128) v2f_kernel(
    const float* __restrict__ X, int xstride, int cin,
    const int* __restrict__ face, const float* __restrict__ W,
    const float* __restrict__ bias, float* __restrict__ out)
{
  __shared__ float At[16 * 64];
  const int rowbase = blockIdx.x * 16;
  const int wave = threadIdx.x >> 5;
  const int KA = 3 * cin;                  // 192 or 576 or 384: multiple of 64
  v8f a0 = {}; v8f a1 = {};
  for (int k0 = 0; k0 < KA; k0 += 64) {
    __syncthreads();
    for (int e = threadIdx.x; e < 16 * 64; e += 128) {
      int r = e >> 6, kk = e & 63, k = k0 + kk;
      int corner = k / cin, c = k - corner * cin;
      int v = face[(size_t)(rowbase + r) * 3 + corner];
      At[e] = X[(size_t)v * xstride + c];
    }
    __syncthreads();
    const float* Wk = W + (size_t)k0 * F4_;
    wmma_chunk64(At, Wk, F4_, wave * 16, a0);
    wmma_chunk64(At, Wk, F4_, (wave + 4) * 16, a1);
  }
  const int lane = threadIdx.x & 31;
  const int c0 = wave * 16 + (lane & 15);
  const int c1 = (wave + 4) * 16 + (lane & 15);
  const int ro = (lane >= 16) ? 8 : 0;
#pragma unroll
  for (int t = 0; t < 8; ++t) {
    int row = rowbase + t + ro;
    out[(size_t)row * F4_ + c0] = a0[t] * (1.f / 3.f) + bias[c0];
    out[(size_t)row * F4_ + c1] = a1[t] * (1.f / 3.f) + bias[c1];
  }
}

// ---------------------------------------------------------------------------
// f2v: agg[v,k9,c] = sum_j coeff[v,j,k9]*mask*F[vtmap[v,j],c]; then GEMM with
// W (1152,O), scaled by 1/max(nf_count,1). O in {128,64}.
// ---------------------------------------------------------------------------
__global__ void __launch_bounds__(128) f2v_kernel(
    const float* __restrict__ F, const int* __restrict__ vtmap,
    const int* __restrict__ nfc, const float* __restrict__ filt,
    const float* __restrict__ W, const float* __restrict__ bias,
    float* __restrict__ out, int O)
{
  __shared__ float At[16 * 64];
  const int rowbase = blockIdx.x * 16;
  const int wave = threadIdx.x >> 5;
  const int KA = K_MESH_ * F4_;            // 1152
  const bool wide = (O == 128);
  v8f a0 = {}; v8f a1 = {};
  for (int k0 = 0; k0 < KA; k0 += 64) {
    __syncthreads();
    for (int e = threadIdx.x; e < 16 * 64; e += 128) {
      int r = e >> 6, kk = e & 63, k = k0 + kk;
      int k9 = k >> 7, c = k & 127;
      int v = rowbase + r;
      int cnt = nfc[v]; cnt = cnt > 1 ? cnt : 1;
      float s = 0.f;
#pragma unroll
      for (int j = 0; j < MAX_NF_; ++j) {
        float msk = (j < cnt) ? 1.f : 0.f;
        int fi = vtmap[(size_t)v * MAX_NF_ + j];
        s += filt[((size_t)v * MAX_NF_ + j) * K_MESH_ + k9] * msk *
             F[(size_t)fi * F4_ + c];
      }
      At[e] = s;
    }
    __syncthreads();
    const float* Wk = W + (size_t)k0 * O;
    wmma_chunk64(At, Wk, O, wave * 16, a0);
    if (wide) wmma_chunk64(At, Wk, O, (wave + 4) * 16, a1);
  }
  const int lane = threadIdx.x & 31;
  const int c0 = wave * 16 + (lane & 15);
  const int c1 = (wave + 4) * 16 + (lane & 15);
  const int ro = (lane >= 16) ? 8 : 0;
#pragma unroll
  for (int t = 0; t < 8; ++t) {
    int row = rowbase + t + ro;
    int cnt = nfc[row]; cnt = cnt > 1 ? cnt : 1;
    float sc = 1.f / (float)cnt;
    out[(size_t)row * O + c0] = a0[t] * sc + bias[c0];
    if (wide) out[(size_t)row * O + c1] = a1[t] * sc + bias[c1];
  }
}

// ---------------------------------------------------------------------------
// dense / transit: plain GEMM rows x (cin -> 128).
// ---------------------------------------------------------------------------
__global__ void __launch_bounds__(128) dense_kernel(
    const float* __restrict__ X, int xstride, int cin,
    const float* __restrict__ W, const float* __restrict__ bias,
    float* __restrict__ out)
{
  __shared__ float At[16 * 64];
  const int rowbase = blockIdx.x * 16;
  const int wave = threadIdx.x >> 5;
  v8f a0 = {}; v8f a1 = {};
  for (int k0 = 0; k0 < cin; k0 += 64) {   // cin in {64,192,320}
    __syncthreads();
    for (int e = threadIdx.x; e < 16 * 64; e += 128) {
      int r = e >> 6, kk = e & 63;
      At[e] = X[(size_t)(rowbase + r) * xstride + k0 + kk];
    }
    __syncthreads();
    const float* Wk = W + (size_t)k0 * F4_;
    wmma_chunk64(At, Wk, F4_, wave * 16, a0);
    wmma_chunk64(At, Wk, F4_, (wave + 4) * 16, a1);
  }
  const int lane = threadIdx.x & 31;
  const int c0 = wave * 16 + (lane & 15);
  const int c1 = (wave + 4) * 16 + (lane & 15);
  const int ro = (lane >= 16) ? 8 : 0;
#pragma unroll
  for (int t = 0; t < 8; ++t) {
    int row = rowbase + t + ro;
    out[(size_t)row * F4_ + c0] = a0[t] + bias[c0];
    out[(size_t)row * F4_ + c1] = a1[t] + bias[c1];
  }
}

// ---------------------------------------------------------------------------
// pconv: agg[n,k,c] = sum_j pt_coeff[n,j,k]*PN[nn_idx[n,j],c]; GEMM with
// W (1280,64), scaled by 1/max(nn_cnt,1).
// ---------------------------------------------------------------------------
__global__ void __launch_bounds__(128) pconv_kernel(
    const float* __restrict__ PN, const int* __restrict__ nnidx,
    const int* __restrict__ nncnt, const float* __restrict__ ptc,
    const float* __restrict__ W, const float* __restrict__ bias,
    float* __restrict__ out)
{
  __shared__ float At[16 * 64];
  const int rowbase = blockIdx.x * 16;
  const int wave = threadIdx.x >> 5;
  const int KA = K_PT_ * F4_;              // 1280
  v8f a0 = {};
  for (int k0 = 0; k0 < KA; k0 += 64) {
    __syncthreads();
    for (int e = threadIdx.x; e < 16 * 64; e += 128) {
      int r = e >> 6, kk = e & 63, k = k0 + kk;
      int k9 = k >> 7, c = k & 127;
      int v = rowbase + r;
      float s = 0.f;
#pragma unroll
      for (int j = 0; j < MAX_NN_; ++j) {
        int nj = nnidx[(size_t)v * MAX_NN_ + j];
        s += ptc[((size_t)v * MAX_NN_ + j) * K_PT_ + k9] *
             PN[(size_t)nj * F4_ + c];
      }
      At[e] = s;
    }
    __syncthreads();
    wmma_chunk64(At, W + (size_t)k0 * G2_, G2_, wave * 16, a0);
  }
  const int lane = threadIdx.x & 31;
  const int c0 = wave * 16 + (lane & 15);
  const int ro = (lane >= 16) ? 8 : 0;
#pragma unroll
  for (int t = 0; t < 8; ++t) {
    int row = rowbase + t + ro;
    int cnt = nncnt[row]; cnt = cnt > 1 ? cnt : 1;
    out[(size_t)row * G2_ + c0] = a0[t] * (1.f / (float)cnt) + bias[c0];
  }
}

// ---------------------------------------------------------------------------
// BatchNorm: per-channel mean / inv-std over rows, then normalize+ReLU.
// ---------------------------------------------------------------------------
__global__ void __launch_bounds__(256) bn_stats_kernel(
    const float* __restrict__ src, int rows, int O, int stride,
    float* __restrict__ stats)
{
  __shared__ float sh[256], sh2[256];
  int c = blockIdx.x, tid = threadIdx.x;
  float s = 0.f, s2 = 0.f;
  for (int r = tid; r < rows; r += 256) {
    float v = src[(size_t)r * stride + c];
    s += v; s2 += v * v;
  }
  sh[tid] = s; sh2[tid] = s2;
  __syncthreads();
  for (int off = 128; off > 0; off >>= 1) {
    if (tid < off) { sh[tid] += sh[tid + off]; sh2[tid] += sh2[tid + off]; }
    __syncthreads();
  }
  if (tid == 0) {
    float inv_n = 1.f / (float)rows;
    float mean = sh[0] * inv_n;
    float var = sh2[0] * inv_n - mean * mean;
    stats[c] = mean;
    stats[128 + c] = rsqrtf(var + EPS_);
  }
}

__global__ void __launch_bounds__(256) bn_apply_kernel(
    const float* __restrict__ src, int rows, int O, int sstride,
    float* __restrict__ dst, int dstride, int doff,
    const float* __restrict__ g, const float* __restrict__ be,
    const float* __restrict__ stats)
{
  int idx = blockIdx.x * blockDim.x + threadIdx.x;   // exact multiple of 256
  int r = idx / O, c = idx - r * O;
  float v = src[(size_t)r * sstride + c];
  float y = (v - stats[c]) * stats[128 + c] * g[c] + be[c];
  dst[(size_t)r * dstride + doff + c] = y > 0.f ? y : 0.f;
}

// ---------------------------------------------------------------------------
// Host launcher
// ---------------------------------------------------------------------------
extern "C" void kernel_launch(void* const* d_in, const int* in_sizes, int n_in,
                              void* d_out, int out_size, void* d_ws,
                              size_t ws_size, hipStream_t stream)
{
  if (n_in < 59) return;  // expect 7 data inputs + 52 param leaves
  const float* inputs = (const float*)d_in[0];
  const float* vertex = (const float*)d_in[1];
  const int*   face   = (const int*)d_in[2];
  const int*   nfc    = (const int*)d_in[3];
  const int*   vtmap  = (const int*)d_in[4];
  const float* filt   = (const float*)d_in[5];
  // d_in[6] = nv_in scalar (NV is compile-time constant here)
  auto P = [&](int i) { return (const float*)d_in[i]; };

  // workspace carve-up (256B aligned)
  char* w = (char*)d_ws;
  auto carve = [&](size_t bytes) {
    char* p = w; w += (bytes + 255) & ~(size_t)255; return p;
  };
  float* xbuf  = (float*)carve((size_t)NV_ * XSTRIDE * 4);
  float* fbuf  = (float*)carve((size_t)NF_ * F4_ * 4);
  float* vA    = (float*)carve((size_t)NV_ * F4_ * 4);
  float* vB    = (float*)carve((size_t)NV_ * F4_ * 4);
  float* v64   = (float*)carve((size_t)NV_ * G2_ * 4);
  float* ptc   = (float*)carve((size_t)NV_ * MAX_NN_ * K_PT_ * 4);
  float* stats = (float*)carve(256 * 4);
  int*   nnidx = (int*)carve((size_t)NV_ * MAX_NN_ * 4);
  int*   nncnt = (int*)carve((size_t)NV_ * 4);

  copy_in_kernel<<<(NV_ * C_IN_) / 256, 256, 0, stream>>>(inputs, xbuf);
  point_graph_kernel<<<(NV_ + 127) / 128, 128, 0, stream>>>(vertex, nnidx,
                                                            nncnt, ptc);

  auto bn = [&](const float* src, int rows, int O, int sstride, float* dst,
                int dstride, int doff, const float* g, const float* be) {
    bn_stats_kernel<<<O, 256, 0, stream>>>(src, rows, O, sstride, stats);
    bn_apply_kernel<<<(rows * O) / 256, 256, 0, stream>>>(
        src, rows, O, sstride, dst, dstride, doff, g, be, stats);
  };

  int cin = C_IN_;
  for (int b = 0; b < 2; ++b) {
    // params pytree leaves (sorted keys): per block
    // dense, m1_f2v, m1_v2f, m2_f2v, m2_v2f, pconv ; each {W,b,be,g}
    int B = 7 + b * 24;
    const float *dW = P(B + 0),  *db = P(B + 1),  *dbe = P(B + 2),  *dg = P(B + 3);
    const float *f1W = P(B + 4), *f1b = P(B + 5), *f1be = P(B + 6), *f1g = P(B + 7);
    const float *v1W = P(B + 8), *v1b = P(B + 9), *v1be = P(B + 10),*v1g = P(B + 11);
    const float *f2W = P(B + 12),*f2b = P(B + 13),*f2be = P(B + 14),*f2g = P(B + 15);
    const float *v2W = P(B + 16),*v2b = P(B + 17),*v2be = P(B + 18),*v2g = P(B + 19);
    const float *pW = P(B + 20), *pb = P(B + 21), *pbe = P(B + 22), *pg = P(B + 23);

    v2f_kernel<<<NF_ / 16, 128, 0, stream>>>(xbuf, XSTRIDE, cin, face, v1W,
                                             v1b, fbuf);
    bn(fbuf, NF_, F4_, F4_, fbuf, F4_, 0, v1g, v1be);
    f2v_kernel<<<NV_ / 16, 128, 0, stream>>>(fbuf, vtmap, nfc, filt, f1W, f1b,
                                             vA, F4_);
    bn(vA, NV_, F4_, F4_, vA, F4_, 0, f1g, f1be);
    v2f_kernel<<<NF_ / 16, 128, 0, stream>>>(vA, F4_, F4_, face, v2W, v2b,
                                             fbuf);
    bn(fbuf, NF_, F4_, F4_, fbuf, F4_, 0, v2g, v2be);
    f2v_kernel<<<NV_ / 16, 128, 0, stream>>>(fbuf, vtmap, nfc, filt, f2W, f2b,
                                             v64, G2_);
    bn(v64, NV_, G2_, G2_, xbuf, XSTRIDE, cin, f2g, f2be);       // concat m
    dense_kernel<<<NV_ / 16, 128, 0, stream>>>(xbuf, XSTRIDE, cin, dW, db, vB);
    bn(vB, NV_, F4_, F4_, vB, F4_, 0, dg, dbe);
    pconv_kernel<<<NV_ / 16, 128, 0, stream>>>(vB, nnidx, nncnt, ptc, pW, pb,
                                               v64);
    bn(v64, NV_, G2_, G2_, xbuf, XSTRIDE, cin + G2_, pg, pbe);   // concat pn
    cin += 2 * G2_;
  }

  // transit: (NV,320) x (320,128) + BN-ReLU -> d_out
  const float *tW = P(55), *tb = P(56), *tbe = P(57), *tg = P(58);
  dense_kernel<<<NV_ / 16, 128, 0, stream>>>(xbuf, XSTRIDE, cin, tW, tb, vA);
  bn(vA, NV_, C_OUT_, C_OUT_, (float*)d_out, C_OUT_, 0, tg, tbe);
}